// GCN_49289044689250
// MI455X (gfx1250) — compile-verified
//
#include <hip/hip_runtime.h>
#include <math.h>

typedef float v2f __attribute__((ext_vector_type(2)));
typedef float v8f __attribute__((ext_vector_type(8)));

#define IN_F 128
#define H_F  64
#define C_F  40

// ---------------------------------------------------------------- degree prep
__global__ void k_init_deg(float* __restrict__ deg, int N) {
  int i = blockIdx.x * blockDim.x + threadIdx.x;
  if (i < N) deg[i] = 1.0f;   // self-loop weight 1
}

__global__ void k_accum_deg(const long long* __restrict__ ei,
                            const float* __restrict__ ew,
                            float* __restrict__ deg, int E) {
  int e = blockIdx.x * blockDim.x + threadIdx.x;
  if (e < E) atomicAdd(&deg[(int)ei[(long long)E + e]], ew[e]);
}

__global__ void k_dinv(const float* __restrict__ deg, float* __restrict__ dinv, int N) {
  int i = blockIdx.x * blockDim.x + threadIdx.x;
  if (i < N) {
    float d = deg[i];
    dinv[i] = (d > 0.0f) ? rsqrtf(d) : 0.0f;
  }
}

// ---------------------------------------------------------------- per-edge metadata:
// norm[e] = dinv[src]*ew*dinv[dst]; also narrow indices to int32 for the hot scatter.
__global__ void k_edge_norm(const long long* __restrict__ ei,
                            const float* __restrict__ ew,
                            const float* __restrict__ dinv,
                            int* __restrict__ src32, int* __restrict__ dst32,
                            float* __restrict__ norm, int E) {
  int e = blockIdx.x * blockDim.x + threadIdx.x;
  if (e >= E) return;
  const int s = (int)ei[e];
  const int d = (int)ei[(long long)E + e];
  src32[e] = s;
  dst32[e] = d;
  norm[e] = dinv[s] * ew[e] * dinv[d];
}

// ---------------------------------------------------------------- GEMM1: h0 = x @ W1  [N,128]x[128,64]
// blockDim = 128 (4 waves), wave w owns 16-col tile w; blockIdx.x owns 16-row tile.
// Also seeds agg1 = h0 * dinv^2 (self-loop term).
__global__ void k_gemm1_wmma(const float* __restrict__ x,
                             const float* __restrict__ W1,
                             const float* __restrict__ dinv,
                             float* __restrict__ h0,
                             float* __restrict__ agg1) {
  const int wave = threadIdx.x >> 5;
  const int lane = threadIdx.x & 31;
  const int half = lane >> 4;     // 0: lanes 0-15, 1: lanes 16-31
  const int l    = lane & 15;
  const int m0   = blockIdx.x * 16;
  const int n0   = wave * 16;

  v8f acc = {};
  const int arow = m0 + l;
  for (int k0 = 0; k0 < IN_F; k0 += 4) {
    const int kb = k0 + 2 * half; // K pair base for this half-wave
    v2f a, b;
    a.x = x[(long long)arow * IN_F + kb];
    a.y = x[(long long)arow * IN_F + kb + 1];
    b.x = W1[(long long)kb * H_F + n0 + l];
    b.y = W1[(long long)(kb + 1) * H_F + n0 + l];
    acc = __builtin_amdgcn_wmma_f32_16x16x4_f32(
        /*neg_a=*/false, a, /*neg_b=*/false, b,
        /*c_mod=*/(short)0, acc, /*reuse_a=*/false, /*reuse_b=*/false);
  }
  // C/D layout: VGPR v -> M = v + 8*half, N = l
#pragma unroll
  for (int v = 0; v < 8; ++v) {
    const int row = m0 + v + 8 * half;
    const int col = n0 + l;
    const float hv = acc[v];
    h0[(long long)row * H_F + col] = hv;
    const float di = dinv[row];
    agg1[(long long)row * H_F + col] = hv * di * di;
  }
}

// ---------------------------------------------------------------- edge scatter, layer 1
// One thread per (edge, 4-feature chunk): 16 chunks/edge. float4 gather + 4 atomics.
__global__ void k_scatter1(const int* __restrict__ src32,
                           const int* __restrict__ dst32,
                           const float* __restrict__ norm,
                           const float* __restrict__ h0,
                           float* __restrict__ agg1,
                           long long total) {
  long long idx = (long long)blockIdx.x * blockDim.x + threadIdx.x;
  if (idx >= total) return;
  const int e = (int)(idx >> 4);        // H_F/4 = 16 chunks
  const int c = (int)(idx & 15) * 4;
  const int s = src32[e];
  const int d = dst32[e];
  const float nm = norm[e];
  const float4 hv = *(const float4*)(h0 + (long long)s * H_F + c);
  float* dst = agg1 + (long long)d * H_F + c;
  atomicAdd(dst + 0, hv.x * nm);
  atomicAdd(dst + 1, hv.y * nm);
  atomicAdd(dst + 2, hv.z * nm);
  atomicAdd(dst + 3, hv.w * nm);
}

// ---------------------------------------------------------------- bias + relu (in place: agg1 -> h), float4
__global__ void k_relu_bias(float4* __restrict__ agg1, const float4* __restrict__ b1v,
                            long long total4) {
  long long idx = (long long)blockIdx.x * blockDim.x + threadIdx.x;
  if (idx >= total4) return;
  const float4 b = b1v[idx & 15];       // 16 float4 chunks per 64-feature row
  float4 v = agg1[idx];
  v.x = fmaxf(v.x + b.x, 0.0f);
  v.y = fmaxf(v.y + b.y, 0.0f);
  v.z = fmaxf(v.z + b.z, 0.0f);
  v.w = fmaxf(v.w + b.w, 0.0f);
  agg1[idx] = v;
}

// ---------------------------------------------------------------- GEMM2: h1 = h @ W2  [N,64]x[64,40]
// 3 waves per block cover cols 0..47 (guard at 40). Seeds agg2 = h1 * dinv^2.
__global__ void k_gemm2_wmma(const float* __restrict__ h,
                             const float* __restrict__ W2,
                             const float* __restrict__ dinv,
                             float* __restrict__ h1,
                             float* __restrict__ agg2) {
  const int wave = threadIdx.x >> 5;   // 0..2
  const int lane = threadIdx.x & 31;
  const int half = lane >> 4;
  const int l    = lane & 15;
  const int m0   = blockIdx.x * 16;
  const int n0   = wave * 16;
  const int col  = n0 + l;
  const bool colOK = (col < C_F);

  v8f acc = {};
  const int arow = m0 + l;
  for (int k0 = 0; k0 < H_F; k0 += 4) {
    const int kb = k0 + 2 * half;
    v2f a, b;
    a.x = h[(long long)arow * H_F + kb];
    a.y = h[(long long)arow * H_F + kb + 1];
    b.x = colOK ? W2[(long long)kb * C_F + col] : 0.0f;
    b.y = colOK ? W2[(long long)(kb + 1) * C_F + col] : 0.0f;
    acc = __builtin_amdgcn_wmma_f32_16x16x4_f32(
        false, a, false, b, (short)0, acc, false, false);
  }
  if (colOK) {
#pragma unroll
    for (int v = 0; v < 8; ++v) {
      const int row = m0 + v + 8 * half;
      const float hv = acc[v];
      h1[(long long)row * C_F + col] = hv;
      const float di = dinv[row];
      agg2[(long long)row * C_F + col] = hv * di * di;
    }
  }
}

// ---------------------------------------------------------------- edge scatter, layer 2
// One thread per (edge, 4-feature chunk): 10 chunks/edge (C_F=40).
__global__ void k_scatter2(const int* __restrict__ src32,
                           const int* __restrict__ dst32,
                           const float* __restrict__ norm,
                           const float* __restrict__ h1,
                           float* __restrict__ agg2,
                           long long total) {
  long long idx = (long long)blockIdx.x * blockDim.x + threadIdx.x;
  if (idx >= total) return;
  const unsigned e = (unsigned)(idx / 10u);
  const unsigned c = (unsigned)(idx % 10u) * 4u;
  const int s = src32[e];
  const int d = dst32[e];
  const float nm = norm[e];
  const float4 hv = *(const float4*)(h1 + (long long)s * C_F + c);
  float* dst = agg2 + (long long)d * C_F + c;
  atomicAdd(dst + 0, hv.x * nm);
  atomicAdd(dst + 1, hv.y * nm);
  atomicAdd(dst + 2, hv.z * nm);
  atomicAdd(dst + 3, hv.w * nm);
}

// ---------------------------------------------------------------- bias + log_softmax (wave per node)
__global__ void k_logsoftmax(const float* __restrict__ agg2,
                             const float* __restrict__ b2,
                             float* __restrict__ out, int N) {
  const int wib  = threadIdx.x >> 5;
  const int lane = threadIdx.x & 31;
  const int n = blockIdx.x * (blockDim.x >> 5) + wib;
  if (n >= N) return;
  const long long base = (long long)n * C_F;
  float v0 = agg2[base + lane] + b2[lane];                 // lane < 32 < 40: always valid
  float v1 = (lane < C_F - 32) ? (agg2[base + 32 + lane] + b2[32 + lane]) : -INFINITY;
  float m = fmaxf(v0, v1);
#pragma unroll
  for (int o = 16; o > 0; o >>= 1) m = fmaxf(m, __shfl_xor(m, o, 32));
  float s = __expf(v0 - m) + ((lane < C_F - 32) ? __expf(v1 - m) : 0.0f);
#pragma unroll
  for (int o = 16; o > 0; o >>= 1) s += __shfl_xor(s, o, 32);
  const float ls = logf(s);
  out[base + lane] = v0 - m - ls;
  if (lane < C_F - 32) out[base + 32 + lane] = v1 - m - ls;
}

// ---------------------------------------------------------------- host launcher
extern "C" void kernel_launch(void* const* d_in, const int* in_sizes, int n_in,
                              void* d_out, int out_size, void* d_ws, size_t ws_size,
                              hipStream_t stream) {
  const float*     x  = (const float*)d_in[0];
  const long long* ei = (const long long*)d_in[1];   // int64 [2,E]
  const float*     ew = (const float*)d_in[2];
  const float*     W1 = (const float*)d_in[3];
  const float*     b1 = (const float*)d_in[4];
  const float*     W2 = (const float*)d_in[5];
  const float*     b2 = (const float*)d_in[6];
  float*           out = (float*)d_out;

  const int N = in_sizes[0] / IN_F;   // 100000
  const int E = in_sizes[2];          // 1600000

  // workspace (floats unless noted):
  // deg[N] | dinv[N] | h0[N*64] | agg1[N*64] | h1[N*40] | norm[E] | src32[E] (int) | dst32[E] (int)
  char* ws = (char*)d_ws;
  float* deg   = (float*)ws;  ws += (size_t)N * sizeof(float);
  float* dinv  = (float*)ws;  ws += (size_t)N * sizeof(float);
  float* h0    = (float*)ws;  ws += (size_t)N * H_F * sizeof(float);
  float* agg1  = (float*)ws;  ws += (size_t)N * H_F * sizeof(float);
  float* h1    = (float*)ws;  ws += (size_t)N * C_F * sizeof(float);
  float* norm  = (float*)ws;  ws += (size_t)E * sizeof(float);
  int*   src32 = (int*)ws;    ws += (size_t)E * sizeof(int);
  int*   dst32 = (int*)ws;    ws += (size_t)E * sizeof(int);
  float* agg2  = h0;          // h0 dead after layer-1 scatter; reuse for layer-2 accumulator

  // 1-3: degree + D^{-1/2}
  k_init_deg <<<(N + 255) / 256, 256, 0, stream>>>(deg, N);
  k_accum_deg<<<(E + 255) / 256, 256, 0, stream>>>(ei, ew, deg, E);
  k_dinv     <<<(N + 255) / 256, 256, 0, stream>>>(deg, dinv, N);

  // 4: per-edge norm + narrowed indices (one pass over int64 edge_index)
  k_edge_norm<<<(E + 255) / 256, 256, 0, stream>>>(ei, ew, dinv, src32, dst32, norm, E);

  // 5: h0 = x@W1, agg1 seeded with self-loop term (N divisible by 16)
  k_gemm1_wmma<<<N / 16, 128, 0, stream>>>(x, W1, dinv, h0, agg1);

  // 6: normalized scatter-add over edges (float4 chunks)
  {
    long long total = (long long)E * (H_F / 4);
    long long blocks = (total + 255) / 256;
    k_scatter1<<<(unsigned)blocks, 256, 0, stream>>>(src32, dst32, norm, h0, agg1, total);
  }

  // 7: h = relu(agg1 + b1), in place, float4
  {
    long long total4 = (long long)N * (H_F / 4);
    long long blocks = (total4 + 255) / 256;
    k_relu_bias<<<(unsigned)blocks, 256, 0, stream>>>((float4*)agg1, (const float4*)b1, total4);
  }

  // 8: h1 = h@W2, agg2 seeded with self-loop term
  k_gemm2_wmma<<<N / 16, 96, 0, stream>>>(agg1, W2, dinv, h1, agg2);

  // 9: layer-2 scatter (float4 chunks)
  {
    long long total = (long long)E * (C_F / 4);
    long long blocks = (total + 255) / 256;
    k_scatter2<<<(unsigned)blocks, 256, 0, stream>>>(src32, dst32, norm, h1, agg2, total);
  }

  // 10: out = log_softmax(agg2 + b2)
  k_logsoftmax<<<(N * 32 + 127) / 128, 128, 0, stream>>>(agg2, b2, out, N);
}